// GCBlock10_23545010717067
// MI455X (gfx1250) — compile-verified
//
#include <hip/hip_runtime.h>
#include <hip/hip_bf16.h>
#include <cstdint>

// ---------------------------------------------------------------------------
// Problem dims (fixed by reference):
//   Q=100, B=4, C=256, H=W=64, C4=1024, C2=512, K9=9, M = B*C = 1024
// ---------------------------------------------------------------------------
#define DQ   100
#define DC   256
#define DC4  1024
#define DM   1024          // B*C
#define HW   4096          // H*W
#define LSTRIDE 68         // LDS row stride; 68*4 = 272 bytes = 17*16 (B128-aligned)

typedef __attribute__((ext_vector_type(2))) float v2f;
typedef __attribute__((ext_vector_type(8))) float v8f;

#if defined(__AMDGCN__) && __has_builtin(__builtin_amdgcn_wmma_f32_16x16x4_f32)
#define HAVE_WMMA_F32 1
#else
#define HAVE_WMMA_F32 0
#endif

// ------------------------- async global->LDS helpers -----------------------
// VDST = per-lane LDS byte address, VADDR = 64-bit global address (VGPR pair).
__device__ __forceinline__ void async_load_b128(unsigned lds_off, float* lds_ptr,
                                                const float* gptr) {
#if defined(__gfx1250__)
  asm volatile("global_load_async_to_lds_b128 %0, %1, off"
               :: "v"(lds_off), "v"(gptr) : "memory");
  (void)lds_ptr;
#else
  lds_ptr[0] = gptr[0]; lds_ptr[1] = gptr[1];
  lds_ptr[2] = gptr[2]; lds_ptr[3] = gptr[3];
#endif
}

__device__ __forceinline__ void wait_async0() {
#if defined(__gfx1250__)
  asm volatile("s_wait_asynccnt 0x0" ::: "memory");
#endif
}

// ---------------------------------------------------------------------------
// K1: pooled[b*C+c] = mean over HxW of value
// ---------------------------------------------------------------------------
__global__ __launch_bounds__(256) void k_pool(const float* __restrict__ value,
                                              float* __restrict__ pooled) {
  __shared__ float red[256];
  const int bc = blockIdx.x;
  const int tid = threadIdx.x;
  const float4* img = (const float4*)(value + (size_t)bc * HW);
  float4 a = img[tid];                  // 256 lanes * float4 = 1024 floats
  float4 b = img[tid + 256];
  float4 c = img[tid + 512];
  float4 d = img[tid + 768];
  float s = (a.x + a.y + a.z + a.w) + (b.x + b.y + b.z + b.w) +
            (c.x + c.y + c.z + c.w) + (d.x + d.y + d.z + d.w);
  red[tid] = s;
  __syncthreads();
  for (int st = 128; st > 0; st >>= 1) {
    if (tid < st) red[tid] += red[tid + st];
    __syncthreads();
  }
  if (tid == 0) pooled[bc] = red[0] * (1.0f / (float)HW);
}

// ---------------------------------------------------------------------------
// K2: dp[b*C4+j] = relu(BN(pooled[b,:] . W1[j,:] + b1[j]))
// ---------------------------------------------------------------------------
__global__ __launch_bounds__(256) void k_dp(const float* __restrict__ pooled,
                                            const float* __restrict__ W1,
                                            const float* __restrict__ b1,
                                            const float* __restrict__ gamma,
                                            const float* __restrict__ beta,
                                            const float* __restrict__ mean,
                                            const float* __restrict__ var,
                                            float* __restrict__ dp) {
  const int id = blockIdx.x * 256 + threadIdx.x;   // 0..4095
  const int b = id >> 10;
  const int j = id & 1023;
  const float4* pr = (const float4*)(pooled + b * DC);
  const float4* wr = (const float4*)(W1 + (size_t)j * DC);
  float s = b1[j];
  for (int c = 0; c < DC / 4; ++c) {
    float4 p = pr[c], w = wr[c];
    s += p.x * w.x + p.y * w.y + p.z * w.z + p.w * w.w;
  }
  s = (s - mean[j]) * rsqrtf(var[j] + 1e-5f) * gamma[j] + beta[j];
  dp[id] = fmaxf(s, 0.f);
}

// ---------------------------------------------------------------------------
// K3: dc = sigmoid(dp @ W2.T + b2); split into former_ch / weight_ch
// ---------------------------------------------------------------------------
__global__ __launch_bounds__(256) void k_dc(const float* __restrict__ dp,
                                            const float* __restrict__ W2,
                                            const float* __restrict__ b2,
                                            float* __restrict__ former_ch,
                                            float* __restrict__ weight_ch) {
  const int id = blockIdx.x * 256 + threadIdx.x;   // 0..2047
  const int b = id >> 9;
  const int i = id & 511;
  const float4* dr = (const float4*)(dp + b * DC4);
  const float4* wr = (const float4*)(W2 + (size_t)i * DC4);
  float s = b2[i];
  for (int j = 0; j < DC4 / 4; ++j) {
    float4 d = dr[j], w = wr[j];
    s += d.x * w.x + d.y * w.y + d.z * w.z + d.w * w.w;
  }
  s = 1.f / (1.f + __expf(-s));
  if (i < DC) former_ch[b * DC + i] = s;
  else        weight_ch[b * DC + (i - DC)] = s;
}

// ---------------------------------------------------------------------------
// K4 (WMMA): lin[m, k] = sum_q query[q*1024+m] * Wl[k*100+q] + bl[k]
//   cf[m*18 + n] = lin * (n<9 ? former_ch[m] : weight_ch[m])
// One wave per 16x16 tile; M=1024 -> 64 tiles; N=18 -> 2 tiles; K=100 (25 steps).
// Branch-free fragment loads: clamp address, select-zero on the value.
// ---------------------------------------------------------------------------
__global__ __launch_bounds__(32) void k_lin(const float* __restrict__ query,
                                            const float* __restrict__ Wl,
                                            const float* __restrict__ bl,
                                            const float* __restrict__ former_ch,
                                            const float* __restrict__ weight_ch,
                                            float* __restrict__ cf) {
  const int m0 = blockIdx.x * 16;
  const int n0 = blockIdx.y * 16;          // 0 or 16
  const int lane = threadIdx.x & 31;
  const int row = lane & 15;
  const int hi  = lane >> 4;               // lane-half -> K offset
  const int n   = n0 + row;
  const bool nv = (n < 18);
  const float* wlp = Wl + (nv ? n : 17) * DQ;   // always-valid row pointer

  v8f acc = {};
#if HAVE_WMMA_F32
  for (int kk = 0; kk < DQ; kk += 4) {     // K=100, exactly 25 steps
    const int ka = kk + hi * 2;
    v2f a, b;
    // A (16x4): lanes 0-15 -> K=kk..kk+1, lanes 16-31 -> K=kk+2..kk+3
    a.x = query[(ka + 0) * DM + m0 + row];
    a.y = query[(ka + 1) * DM + m0 + row];
    // B (4x16): VGPR0 -> K=kk+2*hi, VGPR1 -> K=kk+2*hi+1, N across lanes
    const float t0 = wlp[ka + 0];
    const float t1 = wlp[ka + 1];
    b.x = nv ? t0 : 0.f;
    b.y = nv ? t1 : 0.f;
    acc = __builtin_amdgcn_wmma_f32_16x16x4_f32(false, a, false, b,
                                                (short)0, acc, false, false);
  }
#else
#pragma unroll
  for (int v = 0; v < 8; ++v) {
    const int m = m0 + v + hi * 8;
    float s = 0.f;
    if (nv)
      for (int q = 0; q < DQ; ++q) s += query[q * DM + m] * Wl[n * DQ + q];
    acc[v] = s;
  }
#endif
#pragma unroll
  for (int v = 0; v < 8; ++v) {
    const int m = m0 + v + hi * 8;         // C/D layout: lane-half adds 8 rows
    if (nv) {
      const float lin = acc[v] + bl[n];
      const float ch  = (n < 9) ? former_ch[m] : weight_ch[m];
      cf[m * 18 + n] = lin * ch;
    }
  }
}

// ---------------------------------------------------------------------------
// K5 (WMMA): q_out[n*1024+m] = sum_{k<18} cf[m*18+k]*Wd[n*18+k] + bd[n]
// M=1024 -> 64 tiles; N=100 -> 7 tiles; K=18 padded to 20 (5 steps).
// Output already in (Q, B, C) order: flat index n*1024 + m.
// ---------------------------------------------------------------------------
__global__ __launch_bounds__(32) void k_q(const float* __restrict__ cf,
                                          const float* __restrict__ Wd,
                                          const float* __restrict__ bd,
                                          float* __restrict__ q_out) {
  const int m0 = blockIdx.x * 16;
  const int n0 = blockIdx.y * 16;          // 0,16,...,96
  const int lane = threadIdx.x & 31;
  const int row = lane & 15;
  const int hi  = lane >> 4;
  const int n   = n0 + row;
  const bool nv = (n < DQ);
  const float* ap  = cf + (m0 + row) * 18;
  const float* wdp = Wd + (nv ? n : DQ - 1) * 18;

  v8f acc = {};
#if HAVE_WMMA_F32
#pragma unroll
  for (int kk = 0; kk < 20; kk += 4) {
    const int ka = kk + hi * 2;
    const int k0 = (ka + 0 < 18) ? (ka + 0) : 17;   // folds away except last iter
    const int k1 = (ka + 1 < 18) ? (ka + 1) : 17;
    v2f a, b;
    const float a0 = ap[k0],  a1 = ap[k1];
    const float b0 = wdp[k0], b1 = wdp[k1];
    a.x = (ka + 0 < 18) ? a0 : 0.f;
    a.y = (ka + 1 < 18) ? a1 : 0.f;
    b.x = (nv && ka + 0 < 18) ? b0 : 0.f;
    b.y = (nv && ka + 1 < 18) ? b1 : 0.f;
    acc = __builtin_amdgcn_wmma_f32_16x16x4_f32(false, a, false, b,
                                                (short)0, acc, false, false);
  }
#else
#pragma unroll
  for (int v = 0; v < 8; ++v) {
    const int m = m0 + v + hi * 8;
    float s = 0.f;
    if (nv)
      for (int k = 0; k < 18; ++k) s += cf[m * 18 + k] * Wd[n * 18 + k];
    acc[v] = s;
  }
#endif
#pragma unroll
  for (int v = 0; v < 8; ++v) {
    const int m = m0 + v + hi * 8;
    if (nv) q_out[n * DM + m] = acc[v] + bd[n];
  }
}

// ---------------------------------------------------------------------------
// K6: per-(b,c) 3x3 dynamic stencil. One workgroup per 64x64 image.
// LDS layout: interior pixel (y,x) at lds[(y+1)*68 + x]  (x offset 0 -> the
// B128 async destinations are 16B aligned). Left tap at x=0 wraps into slot 67
// of the previous row; right tap at x=63 hits slot 64. Rows 0/65 and slots
// 64..67 of every interior row are zeroed, so all out-of-range taps read 0,
// matching the reference's zero-padded shifts (Dstack=0 kills those terms).
// ---------------------------------------------------------------------------
__global__ __launch_bounds__(256) void k_stencil(const float* __restrict__ value,
                                                 const float* __restrict__ cf,
                                                 float* __restrict__ y_out) {
  __shared__ __align__(16) float val_lds[66 * LSTRIDE];
  __shared__ __align__(16) float dif_lds[66 * LSTRIDE];
  __shared__ float coef[18];

  const int bc = blockIdx.x;               // 0..1023
  const int tid = threadIdx.x;
  const float* img = value + (size_t)bc * HW;

  if (tid < 18) coef[tid] = cf[bc * 18 + tid];

  // zero the guard cells of both tiles: rows 0 & 65 (68 each) + slots 64..67
  // of rows 1..64 (256)  -> 392 cells per array
  for (int i = tid; i < 392; i += 256) {
    int yy, xx;
    if (i < 68)       { yy = 0;  xx = i; }
    else if (i < 136) { yy = 65; xx = i - 68; }
    else { const int j = i - 136; yy = (j >> 2) + 1; xx = 64 + (j & 3); }
    val_lds[yy * LSTRIDE + xx] = 0.f;
    dif_lds[yy * LSTRIDE + xx] = 0.f;
  }

  // async-copy interior 64x64 -> LDS, 4 pixels (16B) per async op
#pragma unroll
  for (int i = 0; i < 4; ++i) {
    const int seg = tid + i * 256;         // 1024 segments of 4 pixels
    const int p   = seg * 4;
    const int yy  = p >> 6;
    const int xx  = p & 63;                // multiple of 4
    float* lp = &val_lds[(yy + 1) * LSTRIDE + xx];
    const unsigned lds_off = (unsigned)(uintptr_t)lp;  // low 32b = LDS byte offset
    async_load_b128(lds_off, lp, img + p);
  }
  wait_async0();
  __syncthreads();

  // phase 2: y_temp + diff
  float f[9], wkc[9];
#pragma unroll
  for (int k = 0; k < 9; ++k) { f[k] = coef[k]; wkc[k] = coef[9 + k]; }
  const int OFF[9] = { -LSTRIDE - 1, -LSTRIDE, -LSTRIDE + 1,
                       -1, 0, 1,
                       LSTRIDE - 1, LSTRIDE, LSTRIDE + 1 };
  float ytv[16];
#pragma unroll
  for (int i = 0; i < 16; ++i) {
    const int p  = tid + i * 256;
    const int base = ((p >> 6) + 1) * LSTRIDE + (p & 63);
    float yt = 0.f;
#pragma unroll
    for (int k = 0; k < 9; ++k) yt += f[k] * val_lds[base + OFF[k]];
    ytv[i] = yt;
    const float d = val_lds[base] - yt;
    dif_lds[base] = __expf(-d * d);
  }
  __syncthreads();

  // phase 3: y = y_temp - sum_k sigmoid(|W*D - diff*v|) * D * wk
#pragma unroll
  for (int i = 0; i < 16; ++i) {
    const int p  = tid + i * 256;
    const int base = ((p >> 6) + 1) * LSTRIDE + (p & 63);
    const float vC = val_lds[base];
    const float dv = dif_lds[base] * vC;
    float acc = 0.f;
#pragma unroll
    for (int k = 0; k < 9; ++k) {
      const float Dn = val_lds[base + OFF[k]];
      const float Wn = dif_lds[base + OFF[k]];
      const float dw = fabsf(Wn * Dn - dv);
      const float sg = 1.f / (1.f + __expf(-dw));
      acc += sg * Dn * wkc[k];
    }
    y_out[(size_t)bc * HW + p] = ytv[i] - acc;
  }
}

// ---------------------------------------------------------------------------
extern "C" void kernel_launch(void* const* d_in, const int* in_sizes, int n_in,
                              void* d_out, int out_size, void* d_ws, size_t ws_size,
                              hipStream_t stream) {
  (void)in_sizes; (void)n_in; (void)out_size; (void)ws_size;
  const float* query = (const float*)d_in[0];   // (100,4,256)
  const float* value = (const float*)d_in[1];   // (4,256,64,64)
  // d_in[2] hard_sigmoid_masks: unused by the reference
  const float* W1    = (const float*)d_in[3];   // (1024,256)
  const float* b1    = (const float*)d_in[4];
  const float* gmm   = (const float*)d_in[5];
  const float* bet   = (const float*)d_in[6];
  const float* mean  = (const float*)d_in[7];
  const float* var   = (const float*)d_in[8];
  const float* W2    = (const float*)d_in[9];   // (512,1024)
  const float* b2    = (const float*)d_in[10];
  const float* Wl    = (const float*)d_in[11];  // (18,100)
  const float* bl    = (const float*)d_in[12];
  const float* Wd    = (const float*)d_in[13];  // (100,18)
  const float* bd    = (const float*)d_in[14];

  float* ws      = (float*)d_ws;
  float* pooled  = ws;             // 1024
  float* dp      = ws + 1024;      // 4096
  float* fch     = ws + 5120;      // 1024
  float* wch     = ws + 6144;      // 1024
  float* cf      = ws + 7168;      // 1024*18 = 18432  (former|wk interleaved)

  float* q_out = (float*)d_out;            // 100*4*256 = 102400
  float* y_out = q_out + DQ * DM;          // 4*256*64*64

  k_pool   <<<DM, 256, 0, stream>>>(value, pooled);
  k_dp     <<<16, 256, 0, stream>>>(pooled, W1, b1, gmm, bet, mean, var, dp);
  k_dc     <<<8, 256, 0, stream>>>(dp, W2, b2, fch, wch);
  k_lin    <<<dim3(64, 2), 32, 0, stream>>>(query, Wl, bl, fch, wch, cf);
  k_q      <<<dim3(64, 7), 32, 0, stream>>>(cf, Wd, bd, q_out);
  k_stencil<<<DM, 256, 0, stream>>>(value, cf, y_out);
}